// JSDPosLoss_8976481649062
// MI455X (gfx1250) — compile-verified
//
#include <hip/hip_runtime.h>
#include <hip/hip_bf16.h>
#include <math.h>

typedef __attribute__((ext_vector_type(2))) float v2f;
typedef __attribute__((ext_vector_type(8))) float v8f;

#define B_   32
#define HW   4096
#define D_   256
#define NQ   3
#define NPQ  512
#define NPOS 10
#define NPAIR (NPOS * NQ)   // 30 pairs per batch
#define NROW (B_ * NQ)      // 96 attn rows

// ---------------------------------------------------------------------------
// Kernel 1: attn[b][q][j] = dot(sample_z[b][q], z_pos[b][j]) via f32 WMMA.
// Grid (32, 32), block 256 (8 waves). Each wave computes a 16x16 D tile:
// M = query (rows 0..2 valid, rest zero-padded), N = 16 positions.
// ---------------------------------------------------------------------------
__global__ __launch_bounds__(256) void attn_wmma_kernel(
    const float* __restrict__ z, const float* __restrict__ z_pos,
    const int* __restrict__ rand_idx, float* __restrict__ attn) {
  __shared__ float As[16 * D_];  // 16 KB: 16x256 A matrix, rows >= NQ are zero

  const int b   = blockIdx.x;
  const int tid = threadIdx.x;

  // Stage A into LDS: rows 0..2 = gathered sample_z rows, rows 3..15 = 0.
  for (int i = tid; i < 16 * D_; i += 256) {
    int q = i >> 8;            // i / 256
    int c = i & (D_ - 1);
    float v = 0.0f;
    if (q < NQ) {
      int row = rand_idx[b * NQ + q];
      v = z[((size_t)b * HW + row) * D_ + c];
    }
    As[i] = v;
  }
  __syncthreads();

  const int wave = tid >> 5;
  const int lane = tid & 31;
  const int n    = lane & 15;          // column (position within tile) / A row
  const int kofs = (lane >> 4) * 2;    // K sub-offset: 0 for lanes 0-15, 2 for 16-31

  const int tile = blockIdx.y * 8 + wave;   // 0..255
  const int j0   = tile * 16;

  // B fragment source: B[k'][n] = z_pos[b][j0+n][k+k'] -> contiguous pair.
  const float* bp = z_pos + (((size_t)b * HW + j0 + n) * D_) + kofs;
  const float* ap = &As[n * D_ + kofs];

  v8f acc = {};
#pragma unroll 8
  for (int k = 0; k < D_; k += 4) {
    v2f a = *(const v2f*)(ap + k);       // A[M=n][k+kofs], A[M=n][k+kofs+1]
    v2f bb = *(const v2f*)(bp + k);      // B[k+kofs][n],   B[k+kofs+1][n]
    acc = __builtin_amdgcn_wmma_f32_16x16x4_f32(
        /*neg_a=*/false, a, /*neg_b=*/false, bb,
        /*c_mod=*/(short)0, acc, /*reuse_a=*/false, /*reuse_b=*/false);
  }

  // D layout: lanes 0-15 hold M = vgpr index (0..7), N = lane. Queries 0..2
  // live in acc[0..2] of the low half-wave.
  if (lane < 16) {
    attn[((size_t)b * NQ + 0) * HW + j0 + n] = acc[0];
    attn[((size_t)b * NQ + 1) * HW + j0 + n] = acc[1];
    attn[((size_t)b * NQ + 2) * HW + j0 + n] = acc[2];
  }
}

// ---------------------------------------------------------------------------
// Kernel 2: top-10 per attn row (96 rows x 4096). One block per row.
// Ties broken toward the lowest index (matches jax.lax.top_k).
// ---------------------------------------------------------------------------
__global__ __launch_bounds__(256) void topk10_kernel(
    const float* __restrict__ attn, int* __restrict__ topk) {
  __shared__ float vals[HW];   // 16 KB
  __shared__ float sv[256];
  __shared__ int   si[256];

  const int row = blockIdx.x;
  const float* a = attn + (size_t)row * HW;
  for (int i = threadIdx.x; i < HW; i += 256) vals[i] = a[i];
  __syncthreads();

  for (int t = 0; t < NPOS; ++t) {
    float best = -INFINITY;
    int   bidx = HW;
    for (int i = threadIdx.x; i < HW; i += 256) {
      float v = vals[i];
      if (v > best) { best = v; bidx = i; }   // ascending i -> keeps lowest on tie
    }
    sv[threadIdx.x] = best;
    si[threadIdx.x] = bidx;
    __syncthreads();
    for (int s = 128; s > 0; s >>= 1) {
      if (threadIdx.x < (unsigned)s) {
        float ov = sv[threadIdx.x + s];
        int   oi = si[threadIdx.x + s];
        if (ov > sv[threadIdx.x] ||
            (ov == sv[threadIdx.x] && oi < si[threadIdx.x])) {
          sv[threadIdx.x] = ov;
          si[threadIdx.x] = oi;
        }
      }
      __syncthreads();
    }
    if (threadIdx.x == 0) {
      topk[row * NPOS + t] = si[0];
      vals[si[0]] = -INFINITY;   // exclude from later passes
    }
    __syncthreads();
  }
}

// ---------------------------------------------------------------------------
// Kernel 3: JSD. One block per pair row (960). Replicates the reference's
// pairing quirk: within batch b, pair t pairs p=sample_z_dis[b][t%3] with
// sel=z_pos_dis[b][ topk[b][t/10][t%10] ].
// ---------------------------------------------------------------------------
__global__ void zero_out_kernel(float* out) { out[0] = 0.0f; }

__global__ __launch_bounds__(256) void jsd_kernel(
    const float* __restrict__ z_dis, const float* __restrict__ z_pos_dis,
    const int* __restrict__ rand_idx, const int* __restrict__ topk,
    float* __restrict__ out) {
  const int pair = blockIdx.x;          // 0..959
  const int b    = pair / NPAIR;
  const int t    = pair % NPAIR;
  const int qp   = t % NQ;              // query index for p (broadcast quirk)
  const int qs   = t / NPOS;            // query index for selected q
  const int pos  = t % NPOS;

  const int prow = rand_idx[b * NQ + qp];
  const int srow = topk[(b * NQ + qs) * NPOS + pos];

  const float* pp = z_dis     + ((size_t)b * HW + prow) * NPQ;
  const float* sp = z_pos_dis + ((size_t)b * HW + srow) * NPQ;

  float sum = 0.0f;
  for (int c = threadIdx.x; c < NPQ; c += 256) {
    float p  = pp[c];
    float qv = sp[c];
    float m  = 0.5f * (p + qv);
    m = fminf(fmaxf(m, 1e-7f), 1.0f);
    float lm = logf(m);
    float xp = (p  > 0.0f) ? p  * logf(p)  : 0.0f;
    float xq = (qv > 0.0f) ? qv * logf(qv) : 0.0f;
    sum += (xp - p * lm) + (xq - qv * lm);
  }

  __shared__ float red[256];
  red[threadIdx.x] = sum;
  __syncthreads();
  for (int s = 128; s > 0; s >>= 1) {
    if (threadIdx.x < (unsigned)s) red[threadIdx.x] += red[threadIdx.x + s];
    __syncthreads();
  }
  if (threadIdx.x == 0) {
    // loss = 0.5 * (sum_p + sum_q) / n, n = 960 pair rows
    atomicAdd(out, red[0] * (0.5f / (float)(B_ * NPAIR)));
  }
}

// ---------------------------------------------------------------------------
extern "C" void kernel_launch(void* const* d_in, const int* in_sizes, int n_in,
                              void* d_out, int out_size, void* d_ws, size_t ws_size,
                              hipStream_t stream) {
  (void)in_sizes; (void)n_in; (void)out_size; (void)ws_size;
  const float* z         = (const float*)d_in[0];
  const float* z_pos     = (const float*)d_in[1];
  const float* z_dis     = (const float*)d_in[2];
  const float* z_pos_dis = (const float*)d_in[3];
  const int*   rand_idx  = (const int*)d_in[4];

  float* attn = (float*)d_ws;                                // 32*3*4096 f32 = 1.5 MB
  int*   topk = (int*)((char*)d_ws + (size_t)NROW * HW * sizeof(float)); // 960 ints
  float* out  = (float*)d_out;

  zero_out_kernel<<<1, 1, 0, stream>>>(out);
  attn_wmma_kernel<<<dim3(B_, 32), 256, 0, stream>>>(z, z_pos, rand_idx, attn);
  topk10_kernel<<<NROW, 256, 0, stream>>>(attn, topk);
  jsd_kernel<<<B_ * NPAIR, 256, 0, stream>>>(z_dis, z_pos_dis, rand_idx, topk, out);
}